// PairwiseScore_74036646249301
// MI455X (gfx1250) — compile-verified
//
#include <hip/hip_runtime.h>
#include <math.h>

// ---------------------------------------------------------------------------
// Sizes (fixed by the reference problem)
// ---------------------------------------------------------------------------
#define S_    3000
#define P_    80000
#define GI_   768
#define DD_   20
#define HID_  150
#define GIJ_  2364          // 3*GI + 3*DD
#define NT_   10            // hidden padded to 160 = 10 tiles of 16
#define KS1_  591           // 2364 / 4  (K-steps for layer 1, K=4 per WMMA)
#define KS2_  40            // 160  / 4  (K-steps for layer 2)
#define CK_   512           // K-chunk staged in LDS for layer 1
#define ASTR_ 516           // LDS row stride (floats) for A-stage: bank-conflict-free
#define HSTR_ 164           // LDS row stride (floats) for h1/h2 tiles

typedef __attribute__((ext_vector_type(2))) float v2f;
typedef __attribute__((ext_vector_type(8))) float v8f;

__device__ __forceinline__ v8f wmma_f32(v2f a, v2f b, v8f c) {
  // D = A(16x4 f32) * B(4x16 f32) + C(16x16 f32), full fp32 matrix pipe.
  return __builtin_amdgcn_wmma_f32_16x16x4_f32(false, a, false, b, (short)0, c,
                                               false, false);
}

// ---------------------------------------------------------------------------
// Pre-permute W1 (2364x150) and W2 (150x150, padded to 160x160) into the exact
// per-lane B-fragment layout of V_WMMA_F32_16X16X4_F32:
//   lane L supplies B rows k0 = 4*kstep + 2*(L>>4) + {0,1}, column n = nt*16+(L&15)
// One global_load_b64 per lane per WMMA step at GEMM time.
// ---------------------------------------------------------------------------
__global__ void prep_wb(const float* __restrict__ W1, const float* __restrict__ W2,
                        float* __restrict__ W1B, float* __restrict__ W2B) {
  const int N1 = KS1_ * NT_ * 32;
  const int N2 = KS2_ * NT_ * 32;
  int tid = blockIdx.x * blockDim.x + threadIdx.x;
  if (tid < N1) {
    int lane = tid & 31, rest = tid >> 5;
    int nt = rest % NT_, kstep = rest / NT_;
    int n  = nt * 16 + (lane & 15);
    int k0 = 4 * kstep + 2 * (lane >> 4);
    float v0 = 0.f, v1 = 0.f;
    if (n < HID_) { v0 = W1[k0 * HID_ + n]; v1 = W1[(k0 + 1) * HID_ + n]; }
    W1B[2 * tid + 0] = v0;
    W1B[2 * tid + 1] = v1;
  } else if (tid < N1 + N2) {
    int t2 = tid - N1;
    int lane = t2 & 31, rest = t2 >> 5;
    int nt = rest % NT_, kstep = rest / NT_;
    int n  = nt * 16 + (lane & 15);
    int k0 = 4 * kstep + 2 * (lane >> 4);
    float v0 = 0.f, v1 = 0.f;
    if (n < HID_) {
      if (k0 < HID_)     v0 = W2[k0 * HID_ + n];
      if (k0 + 1 < HID_) v1 = W2[(k0 + 1) * HID_ + n];
    }
    W2B[2 * t2 + 0] = v0;
    W2B[2 * t2 + 1] = v1;
  }
}

// mention_ids is sorted: segment s = [lower_bound(s), lower_bound(s+1))
__global__ void seg_bounds(const int* __restrict__ mids, int* __restrict__ segstart) {
  int s = blockIdx.x * blockDim.x + threadIdx.x;
  if (s > S_) return;
  int lo = 0, hi = P_;
  while (lo < hi) {
    int mid = (lo + hi) >> 1;
    if (mids[mid] < s) lo = mid + 1; else hi = mid;
  }
  segstart[s] = lo;
}

// ---------------------------------------------------------------------------
// Fully fused MLP: gather features -> L1 (WMMA, K=2364) -> relu -> L2 (WMMA,
// K=160) -> relu -> L3 dot. One 16-pair tile per 320-thread / 10-wave block;
// wave w owns hidden columns [16w, 16w+16). pairs matrix never materialized.
// ---------------------------------------------------------------------------
__global__ __launch_bounds__(320) void mlp_fused(
    const float* __restrict__ g,    const int* __restrict__ mids,
    const int* __restrict__ aids,   const int* __restrict__ dbins,
    const int* __restrict__ gids,   const int* __restrict__ sids,
    const float* __restrict__ demb, const float* __restrict__ gemb,
    const float* __restrict__ semb, const float* __restrict__ W1B,
    const float* __restrict__ W2B,  const float* __restrict__ b1,
    const float* __restrict__ b2,   const float* __restrict__ W3,
    const float* __restrict__ b3,   float* __restrict__ scores) {
  __shared__ __align__(16) float As[16 * ASTR_];   // 32.25 KB feature chunk
  __shared__ __align__(16) float h1s[16 * HSTR_];  // 10.25 KB
  __shared__ __align__(16) float h2s[16 * HSTR_];  // 10.25 KB
  __shared__ int ii[16], jj[16], db[16], ge[16], sp[16];

  const int t    = threadIdx.x;
  const int lane = t & 31;
  const int wave = t >> 5;           // 0..9 = hidden N-tile
  const int pb   = blockIdx.x * 16;  // base pair index of this tile

  if (t < 16) {
    ii[t] = mids[pb + t];  jj[t] = aids[pb + t];
    db[t] = dbins[pb + t]; ge[t] = gids[pb + t]; sp[t] = sids[pb + t];
  }
  __syncthreads();

  // -------- layer 1: [16 x 2364] @ [2364 x 160] ----------------------------
  v8f acc = {0.f, 0.f, 0.f, 0.f, 0.f, 0.f, 0.f, 0.f};
  const int arow = (lane & 15) * ASTR_ + 2 * (lane >> 4);
  const v2f* W1Bv = (const v2f*)W1B;

  for (int base = 0; base < GIJ_; base += CK_) {
    const int ck = (GIJ_ - base < CK_) ? (GIJ_ - base) : CK_;
    // cooperative, coalesced gather of the 16 x ck feature block
    for (int idx = t; idx < 16 * ck; idx += 320) {
      int p = idx / ck, q = idx - p * ck;
      int qg = base + q;
      float v;
      if (qg < GI_)            v = g[ii[p] * GI_ + qg];
      else if (qg < 2 * GI_)   v = g[jj[p] * GI_ + (qg - GI_)];
      else if (qg < 3 * GI_) { int r = qg - 2 * GI_;
                               v = g[ii[p] * GI_ + r] * g[jj[p] * GI_ + r]; }
      else {
        int r = qg - 3 * GI_;
        if (r < DD_)           v = demb[db[p] * DD_ + r];
        else if (r < 2 * DD_)  v = gemb[ge[p] * DD_ + (r - DD_)];
        else                   v = semb[sp[p] * DD_ + (r - 2 * DD_)];
      }
      As[p * ASTR_ + q] = v;
    }
    __syncthreads();
    const int nks = ck >> 2;
    const int gk0 = base >> 2;
    for (int ks = 0; ks < nks; ++ks) {
      v2f a = *(const v2f*)&As[arow + 4 * ks];                       // ds_load_b64
      v2f b = W1Bv[(size_t)((gk0 + ks) * NT_ + wave) * 32 + lane];   // global b64
      acc = wmma_f32(a, b, acc);
    }
    __syncthreads();
  }

  const int colg  = wave * 16 + (lane & 15);
  const float bi1 = (colg < HID_) ? b1[colg] : 0.f;
#pragma unroll
  for (int r = 0; r < 8; ++r) {
    int pr = r + 8 * (lane >> 4);                    // C layout: M=r / M=r+8
    h1s[pr * HSTR_ + colg] = fmaxf(acc[r] + bi1, 0.f);
  }
  __syncthreads();

  // -------- layer 2: [16 x 160] @ [160 x 160] ------------------------------
  v8f acc2 = {0.f, 0.f, 0.f, 0.f, 0.f, 0.f, 0.f, 0.f};
  const int arow2 = (lane & 15) * HSTR_ + 2 * (lane >> 4);
  const v2f* W2Bv = (const v2f*)W2B;
  for (int ks = 0; ks < KS2_; ++ks) {
    v2f a = *(const v2f*)&h1s[arow2 + 4 * ks];
    v2f b = W2Bv[(ks * NT_ + wave) * 32 + lane];
    acc2 = wmma_f32(a, b, acc2);
  }
  const float bi2 = (colg < HID_) ? b2[colg] : 0.f;
#pragma unroll
  for (int r = 0; r < 8; ++r) {
    int pr = r + 8 * (lane >> 4);
    h2s[pr * HSTR_ + colg] = fmaxf(acc2[r] + bi2, 0.f);
  }
  __syncthreads();

  // -------- layer 3: [16 x 150] @ [150 x 1] --------------------------------
  if (t < 16) {
    float s = b3[0];
    for (int k = 0; k < HID_; ++k) s += h2s[t * HSTR_ + k] * W3[k];
    scores[pb + t] = s;
  }
}

// ---------------------------------------------------------------------------
// Segment-softmax attention: weighted[s] = sum_p attn_p * g[antecedent[p]]
// ---------------------------------------------------------------------------
__global__ __launch_bounds__(256) void attn_weighted(
    const float* __restrict__ g0, const int* __restrict__ aids,
    const float* __restrict__ scores, const int* __restrict__ segstart,
    float* __restrict__ weighted) {
  __shared__ float red[256];
  const int s = blockIdx.x, t = threadIdx.x;
  const int st = segstart[s], en = segstart[s + 1];
  if (en <= st) {
#pragma unroll
    for (int k = 0; k < 3; ++k) weighted[s * GI_ + t + 256 * k] = 0.f;
    return;
  }
  float m = -INFINITY;
  for (int p = st + t; p < en; p += 256) m = fmaxf(m, scores[p]);
  red[t] = m; __syncthreads();
  for (int w = 128; w > 0; w >>= 1) { if (t < w) red[t] = fmaxf(red[t], red[t + w]); __syncthreads(); }
  m = red[0]; __syncthreads();

  float ds = 0.f;
  for (int p = st + t; p < en; p += 256) ds += expf(scores[p] - m);
  red[t] = ds; __syncthreads();
  for (int w = 128; w > 0; w >>= 1) { if (t < w) red[t] += red[t + w]; __syncthreads(); }
  const float inv = 1.f / red[0];

  float a0 = 0.f, a1 = 0.f, a2 = 0.f;
  for (int p = st; p < en; ++p) {
    float a = expf(scores[p] - m) * inv;
    const float* jr = g0 + (size_t)aids[p] * GI_;
    a0 += a * jr[t]; a1 += a * jr[t + 256]; a2 += a * jr[t + 512];
  }
  weighted[s * GI_ + t]       = a0;
  weighted[s * GI_ + t + 256] = a1;
  weighted[s * GI_ + t + 512] = a2;
}

// g1[s] = counts>0 ? f*s + (1-f)*weighted[s] : g0[s]   (f = sigmoid([g0;w]@Wf+bf))
__global__ __launch_bounds__(256) void gate_update(
    const float* __restrict__ g0, const float* __restrict__ weighted,
    const int* __restrict__ segstart, const float* __restrict__ Wf,
    const float* __restrict__ bf, float* __restrict__ g1) {
  __shared__ float red[256];
  __shared__ float fsh;
  const int s = blockIdx.x, t = threadIdx.x;
  const int cnt = segstart[s + 1] - segstart[s];
  if (cnt <= 0) {
#pragma unroll
    for (int k = 0; k < 3; ++k) { int d = t + 256 * k; g1[s * GI_ + d] = g0[s * GI_ + d]; }
    return;
  }
  float part = 0.f;
#pragma unroll
  for (int k = 0; k < 3; ++k) {
    int d = t + 256 * k;
    part += g0[s * GI_ + d] * Wf[d] + weighted[s * GI_ + d] * Wf[GI_ + d];
  }
  red[t] = part; __syncthreads();
  for (int w = 128; w > 0; w >>= 1) { if (t < w) red[t] += red[t + w]; __syncthreads(); }
  if (t == 0) fsh = 1.f / (1.f + expf(-(red[0] + bf[0])));
  __syncthreads();
  const float f = fsh, sf = (float)s;  // reference broadcasts arange(S) here
#pragma unroll
  for (int k = 0; k < 3; ++k) {
    int d = t + 256 * k;
    g1[s * GI_ + d] = f * sf + (1.f - f) * weighted[s * GI_ + d];
  }
}

__global__ void coref_add(const float* __restrict__ mscore, const int* __restrict__ mids,
                          const int* __restrict__ aids, const float* __restrict__ scores,
                          float* __restrict__ out) {
  int p = blockIdx.x * blockDim.x + threadIdx.x;
  if (p >= P_) return;
  out[p] = mscore[mids[p]] + mscore[aids[p]] + scores[p];
}

// final segment softmax with epsilon option; empty segment -> m = 0 via fmaxf(-inf,0)
__global__ __launch_bounds__(128) void final_softmax(
    const float* __restrict__ coref, const int* __restrict__ segstart,
    float* __restrict__ pair_probs, float* __restrict__ eps_probs) {
  __shared__ float red[128];
  const int s = blockIdx.x, t = threadIdx.x;
  const int st = segstart[s], en = segstart[s + 1];
  float m = -INFINITY;
  for (int p = st + t; p < en; p += 128) m = fmaxf(m, coref[p]);
  red[t] = m; __syncthreads();
  for (int w = 64; w > 0; w >>= 1) { if (t < w) red[t] = fmaxf(red[t], red[t + w]); __syncthreads(); }
  m = fmaxf(red[0], 0.f);
  __syncthreads();
  float ds = 0.f;
  for (int p = st + t; p < en; p += 128) ds += expf(coref[p] - m);
  red[t] = ds; __syncthreads();
  for (int w = 64; w > 0; w >>= 1) { if (t < w) red[t] += red[t + w]; __syncthreads(); }
  const float inv = 1.f / (red[0] + expf(-m));
  for (int p = st + t; p < en; p += 128) pair_probs[p] = expf(coref[p] - m) * inv;
  if (t == 0) eps_probs[s] = expf(-m) * inv;
}

// ---------------------------------------------------------------------------
extern "C" void kernel_launch(void* const* d_in, const int* in_sizes, int n_in,
                              void* d_out, int out_size, void* d_ws, size_t ws_size,
                              hipStream_t stream) {
  (void)in_sizes; (void)n_in; (void)out_size; (void)ws_size;
  const float* g_i    = (const float*)d_in[0];
  const float* mscore = (const float*)d_in[1];
  const int*   mids   = (const int*)d_in[2];
  const int*   aids   = (const int*)d_in[3];
  const int*   dbins  = (const int*)d_in[4];
  const int*   gids   = (const int*)d_in[5];
  const int*   sids   = (const int*)d_in[6];
  const float* demb   = (const float*)d_in[7];
  const float* gemb   = (const float*)d_in[8];
  const float* semb   = (const float*)d_in[9];
  const float* W1     = (const float*)d_in[10];
  const float* b1     = (const float*)d_in[11];
  const float* W2     = (const float*)d_in[12];
  const float* b2     = (const float*)d_in[13];
  const float* W3     = (const float*)d_in[14];
  const float* b3     = (const float*)d_in[15];
  const float* Wf     = (const float*)d_in[16];
  const float* bf     = (const float*)d_in[17];

  // workspace layout (~20.4 MB, all fully rewritten every call)
  float* ws       = (float*)d_ws;
  float* W1B      = ws;                                  // 2*KS1*NT*32 = 378240
  float* W2B      = W1B + 2 * KS1_ * NT_ * 32;           // 25600
  float* scores   = W2B + 2 * KS2_ * NT_ * 32;           // P
  float* weighted = scores + P_;                         // S*GI
  float* g1       = weighted + (size_t)S_ * GI_;         // S*GI
  int*   segstart = (int*)(g1 + (size_t)S_ * GI_);       // S+1

  float* out        = (float*)d_out;
  float* coref      = out;            // P
  float* pair_probs = out + P_;       // P
  float* eps_probs  = out + 2 * P_;   // S

  const int NWB = KS1_ * NT_ * 32 + KS2_ * NT_ * 32;
  prep_wb<<<(NWB + 255) / 256, 256, 0, stream>>>(W1, W2, W1B, W2B);
  seg_bounds<<<(S_ + 1 + 255) / 256, 256, 0, stream>>>(mids, segstart);

  // ---- order 0 ----
  mlp_fused<<<P_ / 16, 320, 0, stream>>>(g_i, mids, aids, dbins, gids, sids,
                                         demb, gemb, semb, W1B, W2B,
                                         b1, b2, W3, b3, scores);
  attn_weighted<<<S_, 256, 0, stream>>>(g_i, aids, scores, segstart, weighted);
  gate_update<<<S_, 256, 0, stream>>>(g_i, weighted, segstart, Wf, bf, g1);

  // ---- order 1 ----
  mlp_fused<<<P_ / 16, 320, 0, stream>>>(g1, mids, aids, dbins, gids, sids,
                                         demb, gemb, semb, W1B, W2B,
                                         b1, b2, W3, b3, scores);

  coref_add<<<(P_ + 255) / 256, 256, 0, stream>>>(mscore, mids, aids, scores, coref);
  final_softmax<<<S_, 128, 0, stream>>>(coref, segstart, pair_probs, eps_probs);
}